// JumpingKnowledge_62989990363143
// MI455X (gfx1250) — compile-verified
//
#include <hip/hip_runtime.h>

#define NN 40000
#define NE 640000
#define DIN 128
#define DH 128
#define DOUT 64
#define DCAT (DIN + 2 * DH)   // 384

typedef float v2f __attribute__((ext_vector_type(2)));
typedef float v8f __attribute__((ext_vector_type(8)));

// ---------------- degree / normalization ----------------
__global__ void deg_init_kernel(float* deg, int n) {
    int i = blockIdx.x * blockDim.x + threadIdx.x;
    if (i < n) deg[i] = 1.0f;  // self-loop contributes 1
}

__global__ void deg_accum_kernel(const long long* __restrict__ row, float* deg, int e) {
    int i = blockIdx.x * blockDim.x + threadIdx.x;
    if (i < e) atomicAdd(&deg[(int)row[i]], 1.0f);
}

__global__ void deg_rsqrt_kernel(float* deg, int n) {
    int i = blockIdx.x * blockDim.x + threadIdx.x;
    if (i < n) deg[i] = rsqrtf(deg[i]);  // in-place: deg -> dis
}

// ---------------- dense GEMM via V_WMMA_F32_16X16X4_F32 ----------------
// One wave computes a 32x32 block (2x2 WMMA tiles): per k-step, 2 A-frags +
// 2 B-frags feed 4 WMMAs -> 2 loads per WMMA with full fragment reuse.
__global__ void gemm_wmma_f32_kernel(const float* __restrict__ A,
                                     const float* __restrict__ B,
                                     const float* __restrict__ bias,
                                     float* __restrict__ C,
                                     int M, int N, int K) {
    int wave = blockIdx.x * (blockDim.x >> 5) + (threadIdx.x >> 5);
    int lane = threadIdx.x & 31;
    int nblk = N >> 5;               // 32-wide column blocks
    int tm = wave / nblk;
    int tn = wave - tm * nblk;
    if (tm * 32 >= M) return;

    int ln   = lane & 15;   // M index within A-frag, N index within B/C-frag
    int half = lane >> 4;   // K-pair select (A/B), M-upper-half select (C/D)

    v8f acc[2][2] = {};
    const float* Arow0 = A + (size_t)(tm * 32 + ln) * K;
    const float* Arow1 = Arow0 + (size_t)16 * K;
    const float* Bcol  = B + tn * 32 + ln;

    for (int k = 0; k < K; k += 4) {
        int kk = k + half * 2;
        v2f a0, a1, b0, b1;
        a0.x = Arow0[kk + 0];  a0.y = Arow0[kk + 1];
        a1.x = Arow1[kk + 0];  a1.y = Arow1[kk + 1];
        const float* bp0 = Bcol + (size_t)(kk + 0) * N;
        const float* bp1 = Bcol + (size_t)(kk + 1) * N;
        b0.x = bp0[0];   b0.y = bp1[0];
        b1.x = bp0[16];  b1.y = bp1[16];
        acc[0][0] = __builtin_amdgcn_wmma_f32_16x16x4_f32(false, a0, false, b0,
                                                          (short)0, acc[0][0], false, false);
        acc[0][1] = __builtin_amdgcn_wmma_f32_16x16x4_f32(false, a0, false, b1,
                                                          (short)0, acc[0][1], false, false);
        acc[1][0] = __builtin_amdgcn_wmma_f32_16x16x4_f32(false, a1, false, b0,
                                                          (short)0, acc[1][0], false, false);
        acc[1][1] = __builtin_amdgcn_wmma_f32_16x16x4_f32(false, a1, false, b1,
                                                          (short)0, acc[1][1], false, false);
    }

#pragma unroll
    for (int i = 0; i < 2; ++i) {
#pragma unroll
        for (int j = 0; j < 2; ++j) {
            int n = tn * 32 + j * 16 + ln;
            float bv = bias ? bias[n] : 0.0f;
#pragma unroll
            for (int r = 0; r < 8; ++r) {
                int mm = tm * 32 + i * 16 + r + half * 8;
                C[(size_t)mm * N + n] = acc[i][j][r] + bv;
            }
        }
    }
}

// ---------------- edge gather/scale/scatter-add ----------------
// msg = h[col] * dis[row]*dis[col]; agg[row] += msg. Self-loops as e >= E.
// 32 float4 chunks per edge == exactly one wave32 per edge (coalesced gather).
__global__ void scatter_msg_kernel(const float* __restrict__ h,
                                   const long long* __restrict__ row,
                                   const long long* __restrict__ col,
                                   const float* __restrict__ dis,
                                   float* __restrict__ agg,
                                   int E, int Nn) {
    const int VPT = DH / 4;  // 32
    long long t = (long long)blockIdx.x * blockDim.x + threadIdx.x;
    long long e = t / VPT;
    int f4 = (int)(t - e * VPT);
    if (e >= (long long)E + Nn) return;
    int r, c;
    if (e < E) { r = (int)row[e]; c = (int)col[e]; }
    else       { r = c = (int)(e - E); }
    float w = dis[r] * dis[c];
    const float4 hv = reinterpret_cast<const float4*>(h + (size_t)c * DH)[f4];
    float* dst = agg + (size_t)r * DH + f4 * 4;
    atomicAdd(dst + 0, hv.x * w);
    atomicAdd(dst + 1, hv.y * w);
    atomicAdd(dst + 2, hv.z * w);
    atomicAdd(dst + 3, hv.w * w);
}

// ---------------- bias + LayerNorm + ReLU (one block per node) ----------------
__global__ void bias_ln_relu_kernel(const float* __restrict__ agg,
                                    const float* __restrict__ b,
                                    const float* __restrict__ g,
                                    const float* __restrict__ be,
                                    float* __restrict__ out, int M) {
    __shared__ float red[DH];
    int i = blockIdx.x;
    int f = threadIdx.x;
    float v = agg[(size_t)i * DH + f] + b[f];
    red[f] = v;
    __syncthreads();
    for (int s = DH / 2; s > 0; s >>= 1) {
        if (f < s) red[f] += red[f + s];
        __syncthreads();
    }
    float m = red[0] * (1.0f / DH);
    __syncthreads();
    float d = v - m;
    red[f] = d * d;
    __syncthreads();
    for (int s = DH / 2; s > 0; s >>= 1) {
        if (f < s) red[f] += red[f + s];
        __syncthreads();
    }
    float var = red[0] * (1.0f / DH);
    float y = d * rsqrtf(var + 1e-5f) * g[f] + be[f];
    out[(size_t)i * DH + f] = fmaxf(y, 0.0f);
}

// ---------------- final projection: concat([x,h1,h2]) @ Wp + bp ----------------
// Same 2x2 WMMA blocking; A-fragment gathered from the 3 sources by K-segment
// (segments are 128-aligned; K-steps of 4 never straddle a boundary).
__global__ void proj_wmma_kernel(const float* __restrict__ x,
                                 const float* __restrict__ h1,
                                 const float* __restrict__ h2,
                                 const float* __restrict__ Wp,
                                 const float* __restrict__ bp,
                                 float* __restrict__ out, int M) {
    const int N = DOUT, K = DCAT;
    int wave = blockIdx.x * (blockDim.x >> 5) + (threadIdx.x >> 5);
    int lane = threadIdx.x & 31;
    int nblk = N >> 5;  // 2
    int tm = wave / nblk;
    int tn = wave - tm * nblk;
    if (tm * 32 >= M) return;

    int ln   = lane & 15;
    int half = lane >> 4;
    size_t rowoff0 = (size_t)(tm * 32 + ln) * DH;       // all segments: 128 cols
    size_t rowoff1 = rowoff0 + (size_t)16 * DH;
    const float* Bcol = Wp + tn * 32 + ln;

    v8f acc[2][2] = {};
    for (int k = 0; k < K; k += 4) {
        int kk = k + half * 2;
        const float* seg;
        int kloc;
        if (kk < DIN)           { seg = x;  kloc = kk; }
        else if (kk < DIN + DH) { seg = h1; kloc = kk - DIN; }
        else                    { seg = h2; kloc = kk - DIN - DH; }
        v2f a0, a1, b0, b1;
        a0.x = seg[rowoff0 + kloc + 0];  a0.y = seg[rowoff0 + kloc + 1];
        a1.x = seg[rowoff1 + kloc + 0];  a1.y = seg[rowoff1 + kloc + 1];
        const float* bp0 = Bcol + (size_t)(kk + 0) * N;
        const float* bp1 = Bcol + (size_t)(kk + 1) * N;
        b0.x = bp0[0];   b0.y = bp1[0];
        b1.x = bp0[16];  b1.y = bp1[16];
        acc[0][0] = __builtin_amdgcn_wmma_f32_16x16x4_f32(false, a0, false, b0,
                                                          (short)0, acc[0][0], false, false);
        acc[0][1] = __builtin_amdgcn_wmma_f32_16x16x4_f32(false, a0, false, b1,
                                                          (short)0, acc[0][1], false, false);
        acc[1][0] = __builtin_amdgcn_wmma_f32_16x16x4_f32(false, a1, false, b0,
                                                          (short)0, acc[1][0], false, false);
        acc[1][1] = __builtin_amdgcn_wmma_f32_16x16x4_f32(false, a1, false, b1,
                                                          (short)0, acc[1][1], false, false);
    }

#pragma unroll
    for (int i = 0; i < 2; ++i) {
#pragma unroll
        for (int j = 0; j < 2; ++j) {
            int n = tn * 32 + j * 16 + ln;
            float bv = bp[n];
#pragma unroll
            for (int r = 0; r < 8; ++r) {
                int mm = tm * 32 + i * 16 + r + half * 8;
                out[(size_t)mm * N + n] = acc[i][j][r] + bv;
            }
        }
    }
}

extern "C" void kernel_launch(void* const* d_in, const int* in_sizes, int n_in,
                              void* d_out, int out_size, void* d_ws, size_t ws_size,
                              hipStream_t stream) {
    // setup_inputs order:
    const float*     x   = (const float*)d_in[0];
    const long long* ei  = (const long long*)d_in[1];  // int64 [2, NE]
    const float*     W1  = (const float*)d_in[2];
    const float*     b1  = (const float*)d_in[3];
    const float*     g1  = (const float*)d_in[4];
    const float*     be1 = (const float*)d_in[5];
    const float*     W2  = (const float*)d_in[6];
    const float*     b2  = (const float*)d_in[7];
    const float*     g2  = (const float*)d_in[8];
    const float*     be2 = (const float*)d_in[9];
    const float*     Wp  = (const float*)d_in[10];
    const float*     bp  = (const float*)d_in[11];
    float*           out = (float*)d_out;

    const long long* row = ei;       // edge_index[0] (targets)
    const long long* col = ei + NE;  // edge_index[1] (sources)

    // workspace layout (floats)
    float* ws   = (float*)d_ws;
    float* dis  = ws;                          // NN (deg, then rsqrt in-place)
    float* htmp = dis + NN;                    // NN*DH  (X @ W)
    float* agg  = htmp + (size_t)NN * DH;      // NN*DH  (scatter target)
    float* h1   = agg + (size_t)NN * DH;       // NN*DH
    float* h2   = h1 + (size_t)NN * DH;        // NN*DH

    const int TB = 256;
    // --- degree / dis ---
    deg_init_kernel<<<(NN + TB - 1) / TB, TB, 0, stream>>>(dis, NN);
    deg_accum_kernel<<<(NE + TB - 1) / TB, TB, 0, stream>>>(row, dis, NE);
    deg_rsqrt_kernel<<<(NN + TB - 1) / TB, TB, 0, stream>>>(dis, NN);

    // GEMM grid: one wave per 32x32 block, 8 waves per 256-thread block
    const int blk1 = (NN / 32) * (DH / 32);      // 1250 * 4 = 5000 waves
    const int gemm_blocks = (blk1 + 7) / 8;
    long long scat_threads = ((long long)NE + NN) * (DH / 4);
    const int scat_blocks = (int)((scat_threads + TB - 1) / TB);

    // --- block 1 ---
    gemm_wmma_f32_kernel<<<gemm_blocks, TB, 0, stream>>>(x, W1, nullptr, htmp, NN, DH, DIN);
    hipMemsetAsync(agg, 0, (size_t)NN * DH * sizeof(float), stream);
    scatter_msg_kernel<<<scat_blocks, TB, 0, stream>>>(htmp, row, col, dis, agg, NE, NN);
    bias_ln_relu_kernel<<<NN, DH, 0, stream>>>(agg, b1, g1, be1, h1, NN);

    // --- block 2 ---
    gemm_wmma_f32_kernel<<<gemm_blocks, TB, 0, stream>>>(h1, W2, nullptr, htmp, NN, DH, DH);
    hipMemsetAsync(agg, 0, (size_t)NN * DH * sizeof(float), stream);
    scatter_msg_kernel<<<scat_blocks, TB, 0, stream>>>(htmp, row, col, dis, agg, NE, NN);
    bias_ln_relu_kernel<<<NN, DH, 0, stream>>>(agg, b2, g2, be2, h2, NN);

    // --- jumping-knowledge projection ---
    const int blkP = (NN / 32) * (DOUT / 32);    // 1250 * 2 = 2500 waves
    proj_wmma_kernel<<<(blkP + 7) / 8, TB, 0, stream>>>(x, h1, h2, Wp, bp, out, NN);
}